// GPSLayer_14955076124866
// MI455X (gfx1250) — compile-verified
//
#include <hip/hip_runtime.h>

typedef __attribute__((ext_vector_type(2))) float v2f;
typedef __attribute__((ext_vector_type(8))) float v8f;

#define IN_CH   128
#define OUT_CH  64
#define POS_DIM 64

// ---------------------------------------------------------------------------
// K1: deg[i] = 1 (self loop)
// ---------------------------------------------------------------------------
__global__ void init_deg_kernel(float* __restrict__ deg, int N) {
    int i = blockIdx.x * blockDim.x + threadIdx.x;
    if (i < N) deg[i] = 1.0f;
}

// K2: deg[dst[e]] += 1 for each real edge
__global__ void count_deg_kernel(const int* __restrict__ dst, float* __restrict__ deg, int E) {
    int e = blockIdx.x * blockDim.x + threadIdx.x;
    if (e < E) atomicAdd(&deg[dst[e]], 1.0f);
}

// K3: dinv[i] = rsqrt(deg[i])   (deg >= 1 always due to self loop)
__global__ void rsqrt_kernel(float* __restrict__ deg, int N) {
    int i = blockIdx.x * blockDim.x + threadIdx.x;
    if (i < N) deg[i] = rsqrtf(deg[i]);
}

// ---------------------------------------------------------------------------
// K4: h = x @ W  via V_WMMA_F32_16X16X4_F32.
// One wave computes one 16(M) x 16(N) fp32 tile; K-loop of 32 WMMAs.
// A frag (16x4 f32): lane l -> m = l%16, holds A[m][2*(l/16)+0..1]  (2 VGPRs)
// B frag (4x16 f32): lane l -> n = l%16, holds B[2*(l/16)+0..1][n]
// C/D (16x16 f32):   lane l -> n = l%16, vgpr r -> m = (l/16)*8 + r
// ---------------------------------------------------------------------------
__global__ void gemm_xw_kernel(const float* __restrict__ x,
                               const float* __restrict__ W,
                               float* __restrict__ h, int N, int mTiles) {
    const int wavesPerBlock = blockDim.x >> 5;
    const int waveId = threadIdx.x >> 5;
    const int lane   = threadIdx.x & 31;
    const int mTile  = blockIdx.x * wavesPerBlock + waveId;
    const int nTile  = blockIdx.y;
    if (mTile >= mTiles) return;                 // wave-uniform exit: EXEC stays all-1s

    const int l16 = lane & 15;
    const int kh  = (lane >> 4) << 1;            // 0 or 2
    int m = mTile * 16 + l16;
    if (m >= N) m = N - 1;                       // clamp loads only (keep EXEC full)
    const int n = nTile * 16 + l16;

    v8f c = {};
#pragma unroll
    for (int k = 0; k < IN_CH; k += 4) {
        v2f a = *(const v2f*)(x + (size_t)m * IN_CH + k + kh);
        v2f bfrag;
        bfrag[0] = W[(size_t)(k + kh + 0) * OUT_CH + n];
        bfrag[1] = W[(size_t)(k + kh + 1) * OUT_CH + n];
        c = __builtin_amdgcn_wmma_f32_16x16x4_f32(
                false, a, false, bfrag, (short)0, c, false, false);
    }

    const int rbase = mTile * 16 + ((lane >> 4) << 3);
#pragma unroll
    for (int r = 0; r < 8; ++r) {
        int row = rbase + r;
        if (row < N) h[(size_t)row * OUT_CH + n] = c[r];
    }
}

// ---------------------------------------------------------------------------
// K5: out = pos @ Wp + bp + b + h * dinv^2   (self-loop term fused here)
// Same WMMA tiling, K = 64 -> 16 WMMAs per tile. Writes every out element.
// ---------------------------------------------------------------------------
__global__ void gemm_pos_fused_kernel(const float* __restrict__ pos,
                                      const float* __restrict__ Wp,
                                      const float* __restrict__ bp,
                                      const float* __restrict__ b,
                                      const float* __restrict__ h,
                                      const float* __restrict__ dinv,
                                      float* __restrict__ out, int N, int mTiles) {
    const int wavesPerBlock = blockDim.x >> 5;
    const int waveId = threadIdx.x >> 5;
    const int lane   = threadIdx.x & 31;
    const int mTile  = blockIdx.x * wavesPerBlock + waveId;
    const int nTile  = blockIdx.y;
    if (mTile >= mTiles) return;

    const int l16 = lane & 15;
    const int kh  = (lane >> 4) << 1;
    int m = mTile * 16 + l16;
    if (m >= N) m = N - 1;
    const int n = nTile * 16 + l16;

    v8f c = {};
#pragma unroll
    for (int k = 0; k < POS_DIM; k += 4) {
        v2f a = *(const v2f*)(pos + (size_t)m * POS_DIM + k + kh);
        v2f bfrag;
        bfrag[0] = Wp[(size_t)(k + kh + 0) * OUT_CH + n];
        bfrag[1] = Wp[(size_t)(k + kh + 1) * OUT_CH + n];
        c = __builtin_amdgcn_wmma_f32_16x16x4_f32(
                false, a, false, bfrag, (short)0, c, false, false);
    }

    const float bias = bp[n] + b[n];
    const int rbase = mTile * 16 + ((lane >> 4) << 3);
#pragma unroll
    for (int r = 0; r < 8; ++r) {
        int row = rbase + r;
        if (row < N) {
            float di = dinv[row];
            out[(size_t)row * OUT_CH + n] =
                c[r] + bias + h[(size_t)row * OUT_CH + n] * di * di;
        }
    }
}

// ---------------------------------------------------------------------------
// K6: edge scatter. One wave per edge; each lane handles 2 channels.
// out[dst] += h[src] * dinv[src]*dinv[dst]   (global_atomic_add_f32, L2-resident)
// ---------------------------------------------------------------------------
__global__ void edge_scatter_kernel(const int* __restrict__ src,
                                    const int* __restrict__ dst,
                                    const float* __restrict__ h,
                                    const float* __restrict__ dinv,
                                    float* __restrict__ out, int E) {
    const int wavesPerBlock = blockDim.x >> 5;
    const int e    = blockIdx.x * wavesPerBlock + (threadIdx.x >> 5);
    const int lane = threadIdx.x & 31;
    if (e >= E) return;

    const int s = src[e];
    const int d = dst[e];
    const float nrm = dinv[s] * dinv[d];

    const float2 hv = *(const float2*)(h + (size_t)s * OUT_CH + 2 * lane);
    float* op = out + (size_t)d * OUT_CH + 2 * lane;
    atomicAdd(op + 0, hv.x * nrm);
    atomicAdd(op + 1, hv.y * nrm);
}

// ---------------------------------------------------------------------------
// Host-side launcher
// Inputs: 0=x[N,128] 1=edge_index[2,E] 2=pos[N,64] 3=W[128,64] 4=b[64]
//         5=Wp[64,64] 6=bp[64]
// Workspace: dinv (N floats) | h (N*64 floats)  -> ~26 MB
// ---------------------------------------------------------------------------
extern "C" void kernel_launch(void* const* d_in, const int* in_sizes, int n_in,
                              void* d_out, int out_size, void* d_ws, size_t ws_size,
                              hipStream_t stream) {
    const float* x   = (const float*)d_in[0];
    const int*   ei  = (const int*)d_in[1];
    const float* pos = (const float*)d_in[2];
    const float* W   = (const float*)d_in[3];
    const float* b   = (const float*)d_in[4];
    const float* Wp  = (const float*)d_in[5];
    const float* bp  = (const float*)d_in[6];
    float* out = (float*)d_out;

    const int N = in_sizes[0] / IN_CH;     // 100000
    const int E = in_sizes[1] / 2;         // 1600000
    const int* src = ei;
    const int* dst = ei + E;

    float* dinv = (float*)d_ws;            // N floats (holds deg, then rsqrt in place)
    float* h    = dinv + ((N + 3) & ~3);   // N*OUT_CH floats

    // 1) degree of A + I
    init_deg_kernel<<<(N + 255) / 256, 256, 0, stream>>>(dinv, N);
    count_deg_kernel<<<(E + 255) / 256, 256, 0, stream>>>(dst, dinv, E);
    rsqrt_kernel<<<(N + 255) / 256, 256, 0, stream>>>(dinv, N);

    // 2) h = x @ W  (fp32 WMMA)
    const int mTiles = (N + 15) / 16;                       // 6250
    const int wavesPerBlock = 8;                            // 256 threads
    dim3 gGemm((mTiles + wavesPerBlock - 1) / wavesPerBlock, OUT_CH / 16);
    gemm_xw_kernel<<<gGemm, 32 * wavesPerBlock, 0, stream>>>(x, W, h, N, mTiles);

    // 3) out = pos @ Wp + bp + b + h * dinv^2   (initializes every out element)
    gemm_pos_fused_kernel<<<gGemm, 32 * wavesPerBlock, 0, stream>>>(
        pos, Wp, bp, b, h, dinv, out, N, mTiles);

    // 4) scatter-add over real edges
    edge_scatter_kernel<<<(E + wavesPerBlock - 1) / wavesPerBlock,
                          32 * wavesPerBlock, 0, stream>>>(src, dst, h, dinv, out, E);
}